// MolecularGNN_19370302505652
// MI455X (gfx1250) — compile-verified
//
#include <hip/hip_runtime.h>
#include <hip/hip_bf16.h>

// ---------------------------------------------------------------------------
// MolecularGNN on MI455X (gfx1250, wave32).
// Algebraic collapse: edge-encoder + edge-linear + attention-edge-dot fold to
// a 3x16 per-edge matmul (precomputed once for all 4 layers). The remaining
// GEMM (h @ lin_w, f32) uses V_WMMA_F32_16X16X4_F32, with the A tile staged
// into LDS via the CDNA5 async Global->LDS engine (ASYNCcnt).
// ---------------------------------------------------------------------------

#define NN   50000
#define EE   640000
#define GG   512
#define HIDD 128
#define ET   (EE + NN)   // edges + self loops

typedef __attribute__((ext_vector_type(2))) float v2f;
typedef __attribute__((ext_vector_type(8))) float v8f;

__device__ __forceinline__ void atomAddF(float* p, float v) {
  __hip_atomic_fetch_add(p, v, __ATOMIC_RELAXED, __HIP_MEMORY_SCOPE_AGENT);
}
__device__ __forceinline__ void atomMaxU(unsigned* p, unsigned v) {
  __hip_atomic_fetch_max(p, v, __ATOMIC_RELAXED, __HIP_MEMORY_SCOPE_AGENT);
}
// order-preserving float->uint map (monotone for all non-NaN), inverse below
__device__ __forceinline__ unsigned mapf(float f) {
  unsigned b = __float_as_uint(f);
  return (b & 0x80000000u) ? ~b : (b | 0x80000000u);
}
__device__ __forceinline__ float unmapf(unsigned u) {
  return __uint_as_float((u & 0x80000000u) ? (u & 0x7FFFFFFFu) : ~u);
}

// ---------------- node encoder: h = x(8) @ W(8x128) + b --------------------
__global__ __launch_bounds__(256) void enc_node_kernel(
    const float* __restrict__ x, const float* __restrict__ w,
    const float* __restrict__ b, float* __restrict__ h) {
  int idx = blockIdx.x * 256 + threadIdx.x;
  if (idx >= NN * HIDD) return;
  int node = idx >> 7, c = idx & 127;
  float acc = b[c];
#pragma unroll
  for (int f = 0; f < 8; ++f) acc += x[node * 8 + f] * w[f * HIDD + c];
  h[idx] = acc;
}

// ---------------- per-dst degree + sum of raw 3 edge features --------------
__global__ __launch_bounds__(256) void edge_prep_kernel(
    const int* __restrict__ dst, const float* __restrict__ attr,
    float* __restrict__ attrsum, float* __restrict__ deg) {
  int t = blockIdx.x * 256 + threadIdx.x;
  if (t >= EE) return;
  int d = dst[t];
  atomAddF(&deg[d], 1.0f);
#pragma unroll
  for (int f = 0; f < 3; ++f) atomAddF(&attrsum[d * 3 + f], attr[t * 3 + f]);
}

// --------- fold lin_e_w x a_edge -> Weff(128x16); then W2(3x16), b2(16) ----
__global__ __launch_bounds__(256) void weff_kernel(
    const float* __restrict__ lin_e_w,   // [4][128][128]
    const float* __restrict__ a_edge,    // [4][4][32]
    const float* __restrict__ enc_e_w,   // [3][128]
    const float* __restrict__ enc_e_b,   // [128]
    float* __restrict__ W2, float* __restrict__ b2) {
  __shared__ float Weff[HIDD * 16];
  for (int i = threadIdx.x; i < HIDD * 16; i += 256) {
    int k = i >> 4, j = i & 15;
    int l = j >> 2, hh = j & 3;
    float s = 0.f;
    const float* wrow = lin_e_w + ((size_t)l * HIDD + k) * HIDD + hh * 32;
    const float* arow = a_edge + (l * 4 + hh) * 32;
#pragma unroll 8
    for (int c = 0; c < 32; ++c) s += wrow[c] * arow[c];
    Weff[k * 16 + j] = s;
  }
  __syncthreads();
  for (int i = threadIdx.x; i < 64; i += 256) {
    if (i < 48) {
      int f = i >> 4, j = i & 15;
      float s = 0.f;
      for (int k = 0; k < HIDD; ++k) s += enc_e_w[f * HIDD + k] * Weff[k * 16 + j];
      W2[f * 16 + j] = s;
    } else {
      int j = i - 48;
      float s = 0.f;
      for (int k = 0; k < HIDD; ++k) s += enc_e_b[k] * Weff[k * 16 + j];
      b2[j] = s;
    }
  }
}

// --------- e for real edges (all 4 layers x 4 heads at once) --------------
__global__ __launch_bounds__(256) void e_real_kernel(
    const float* __restrict__ attr, const float* __restrict__ W2,
    const float* __restrict__ b2, float* __restrict__ e_real) {
  int t = blockIdx.x * 256 + threadIdx.x;
  if (t >= EE) return;
  float a0 = attr[t * 3 + 0], a1 = attr[t * 3 + 1], a2 = attr[t * 3 + 2];
#pragma unroll
  for (int j = 0; j < 16; ++j)
    e_real[(size_t)t * 16 + j] =
        b2[j] + a0 * W2[j] + a1 * W2[16 + j] + a2 * W2[32 + j];
}

// --------- e for self-loop edges (fill_value='mean' folded) ---------------
__global__ __launch_bounds__(256) void e_loop_kernel(
    const float* __restrict__ attrsum, const float* __restrict__ deg,
    const float* __restrict__ W2, const float* __restrict__ b2,
    float* __restrict__ e_loop) {
  int n = blockIdx.x * 256 + threadIdx.x;
  if (n >= NN) return;
  float dg = deg[n];
  float inv = 1.0f / fmaxf(dg, 1.0f);
  float a0 = attrsum[n * 3 + 0], a1 = attrsum[n * 3 + 1], a2 = attrsum[n * 3 + 2];
#pragma unroll
  for (int j = 0; j < 16; ++j)
    e_loop[(size_t)n * 16 + j] =
        (b2[j] * dg + a0 * W2[j] + a1 * W2[16 + j] + a2 * W2[32 + j]) * inv;
}

// ---------------- xp = h @ lin_w via V_WMMA_F32_16X16X4_F32 ----------------
// block = 256 thr = 8 waves; block covers 16 rows; wave w -> cols [16w,16w+16)
// A tile staged in LDS by async Global->LDS DMA (no VGPR round trip), rows
// padded to 132 floats (528B) so fragment reads are bank-conflict-free.
__global__ __launch_bounds__(256) void xp_gemm_wmma(
    const float* __restrict__ Hm, const float* __restrict__ W,
    float* __restrict__ XP) {
  __shared__ float As[16 * 132];
  const int row0 = blockIdx.x * 16;
  const int tid = threadIdx.x;
  // 16 rows x 128 floats = 512 x b128 chunks (32 chunks per row), 2 per thread
  const unsigned lds_base = (unsigned)(size_t)(&As[0]);
#pragma unroll
  for (int i = tid; i < 512; i += 256) {
    int r = i >> 5, c16 = i & 31;
    unsigned lds = lds_base + r * 528 + c16 * 16;
    const float* gp = Hm + (size_t)(row0 + r) * HIDD + c16 * 4;
    asm volatile("global_load_async_to_lds_b128 %0, %1, off"
                 :: "v"(lds), "v"((unsigned long long)(size_t)gp)
                 : "memory");
  }
  asm volatile("s_wait_asynccnt 0x0" ::: "memory");
  __syncthreads();
  const int lane = tid & 31;
  const int wave = tid >> 5;
  const int col0 = wave * 16;
  const int lm = lane & 15;
  const bool hi = lane >= 16;
  v8f acc = {};
#pragma unroll 8
  for (int k0 = 0; k0 < HIDD; k0 += 4) {
    int ka = k0 + (hi ? 2 : 0);
    v2f a, b;
    // A 16x4 frag: lanes0-15 rows M=0..15 K={k0,k0+1}; lanes16-31 K={k0+2,k0+3}
    a.x = As[lm * 132 + ka];
    a.y = As[lm * 132 + ka + 1];
    // B 4x16 frag: VGPR r holds rows K=r (lanes0-15) / K=r+2 (lanes16-31)
    b.x = W[(size_t)ka * HIDD + col0 + lm];
    b.y = W[(size_t)(ka + 1) * HIDD + col0 + lm];
    acc = __builtin_amdgcn_wmma_f32_16x16x4_f32(
        /*neg_a=*/false, a, /*neg_b=*/false, b,
        /*c_mod=*/(short)0, acc, /*reuse_a=*/false, /*reuse_b=*/false);
  }
  // C layout: VGPR r -> M=r (lanes0-15) / M=r+8 (lanes16-31), N = lane&15
#pragma unroll
  for (int r = 0; r < 8; ++r) {
    int row = row0 + r + (hi ? 8 : 0);
    XP[(size_t)row * HIDD + col0 + lm] = acc[r];
  }
}

// ---------------- s,d per (node, head) from xp ----------------------------
__global__ __launch_bounds__(256) void sd_kernel(
    const float* __restrict__ xp, const float* __restrict__ asrc,
    const float* __restrict__ adst, float* __restrict__ s,
    float* __restrict__ d) {
  int i = blockIdx.x * 256 + threadIdx.x;
  if (i >= NN * 4) return;
  int n = i >> 2, hh = i & 3;
  const float* xr = xp + (size_t)n * HIDD + hh * 32;
  const float* as = asrc + hh * 32;
  const float* ad = adst + hh * 32;
  float ss = 0.f, dd = 0.f;
#pragma unroll 8
  for (int c = 0; c < 32; ++c) {
    float v = xr[c];
    ss += v * as[c];
    dd += v * ad[c];
  }
  s[i] = ss;
  d[i] = dd;
}

__device__ __forceinline__ void edge_ids(int t, const int* src, const int* dst,
                                         int& st, int& dt) {
  if (t < EE) { st = src[t]; dt = dst[t]; }
  else { st = t - EE; dt = t - EE; }
}

// ---------------- attention pass 1: segment max (mapped uint) -------------
__global__ __launch_bounds__(256) void att_pass1(
    const int* __restrict__ src, const int* __restrict__ dst,
    const float* __restrict__ s, const float* __restrict__ d,
    const float* __restrict__ e_real, const float* __restrict__ e_loop,
    unsigned* __restrict__ m, int layerOff) {
  int t = blockIdx.x * 256 + threadIdx.x;
  if (t >= ET) return;
  int st, dt;
  edge_ids(t, src, dst, st, dt);
  const float* ev = (t < EE) ? (e_real + (size_t)t * 16 + layerOff)
                             : (e_loop + (size_t)(t - EE) * 16 + layerOff);
#pragma unroll
  for (int hh = 0; hh < 4; ++hh) {
    float a = s[st * 4 + hh] + d[dt * 4 + hh] + ev[hh];
    a = (a > 0.f) ? a : 0.2f * a;  // leaky_relu 0.2
    atomMaxU(&m[dt * 4 + hh], mapf(a));
  }
}

// ---------------- attention pass 2: exp + segment sum ---------------------
__global__ __launch_bounds__(256) void att_pass2(
    const int* __restrict__ src, const int* __restrict__ dst,
    const float* __restrict__ s, const float* __restrict__ d,
    const float* __restrict__ e_real, const float* __restrict__ e_loop,
    const unsigned* __restrict__ m, float* __restrict__ wbuf,
    float* __restrict__ z, int layerOff) {
  int t = blockIdx.x * 256 + threadIdx.x;
  if (t >= ET) return;
  int st, dt;
  edge_ids(t, src, dst, st, dt);
  const float* ev = (t < EE) ? (e_real + (size_t)t * 16 + layerOff)
                             : (e_loop + (size_t)(t - EE) * 16 + layerOff);
#pragma unroll
  for (int hh = 0; hh < 4; ++hh) {
    float a = s[st * 4 + hh] + d[dt * 4 + hh] + ev[hh];
    a = (a > 0.f) ? a : 0.2f * a;
    float w = __expf(a - unmapf(m[dt * 4 + hh]));
    wbuf[(size_t)t * 4 + hh] = w;
    atomAddF(&z[dt * 4 + hh], w);
  }
}

// ---------------- attention pass 3: weighted message scatter --------------
// 256 threads = 2 edges x 128 channels; xp + accumulator are L2-resident.
__global__ __launch_bounds__(256) void att_pass3(
    const int* __restrict__ src, const int* __restrict__ dst,
    const float* __restrict__ xp, const float* __restrict__ wbuf,
    const float* __restrict__ z, float* __restrict__ outacc) {
  int t = blockIdx.x * 2 + (threadIdx.x >> 7);
  if (t >= ET) return;
  int ch = threadIdx.x & 127;
  int hh = ch >> 5;
  int st, dt;
  edge_ids(t, src, dst, st, dt);
  float w = wbuf[(size_t)t * 4 + hh];
  float coef = w / (z[dt * 4 + hh] + 1e-16f);
  atomAddF(&outacc[(size_t)dt * HIDD + ch],
           xp[(size_t)st * HIDD + ch] * coef);
}

// ---------------- batchnorm stats (sum, sumsq of hn+bias) -----------------
__global__ __launch_bounds__(256) void bn_stats(
    const float* __restrict__ hn, const float* __restrict__ bias,
    float* __restrict__ sums) {  // sums[0:128]=sum, [128:256]=sumsq
  int c = threadIdx.x & 127;
  int half = threadIdx.x >> 7;
  float bc = bias[c];
  float s = 0.f, q = 0.f;
  for (int node = blockIdx.x * 2 + half; node < NN; node += gridDim.x * 2) {
    float v = hn[(size_t)node * HIDD + c] + bc;
    s += v;
    q += v * v;
  }
  atomAddF(&sums[c], s);
  atomAddF(&sums[128 + c], q);
}

// --------- batchnorm apply + relu + residual into h -----------------------
__global__ __launch_bounds__(256) void bn_apply(
    const float* __restrict__ hn, const float* __restrict__ bias,
    const float* __restrict__ sums, const float* __restrict__ gamma,
    const float* __restrict__ beta, float* __restrict__ h, int layer) {
  int idx = blockIdx.x * 256 + threadIdx.x;
  if (idx >= NN * HIDD) return;
  int c = idx & 127;
  const float invN = 1.0f / (float)NN;
  float mu = sums[c] * invN;
  float var = sums[128 + c] * invN - mu * mu;
  float v = hn[idx] + bias[c];
  v = (v - mu) * rsqrtf(var + 1e-5f) * gamma[c] + beta[c];
  v = fmaxf(v, 0.f);
  h[idx] = layer ? (h[idx] + v) : v;
}

// ---------------- global mean pool accumulation ---------------------------
__global__ __launch_bounds__(256) void pool_kernel(
    const float* __restrict__ h, const int* __restrict__ batch,
    float* __restrict__ pool, float* __restrict__ cnt) {
  int idx = blockIdx.x * 256 + threadIdx.x;
  if (idx >= NN * HIDD) return;
  int node = idx >> 7, c = idx & 127;
  int g = batch[node];
  atomAddF(&pool[(size_t)g * HIDD + c], h[idx]);
  if (c == 0) atomAddF(&cnt[g], 1.0f);
}

// ---------------- FC head: 512 graphs, one block each ---------------------
__global__ __launch_bounds__(128) void head_kernel(
    const float* __restrict__ pool, const float* __restrict__ cnt,
    const float* __restrict__ fc1w, const float* __restrict__ fc1b,
    const float* __restrict__ fc2w, const float* __restrict__ fc2b,
    const float* __restrict__ fc3w, const float* __restrict__ fc3b,
    float* __restrict__ out) {
  __shared__ float g[128], t1[128], t2[64];
  int gi = blockIdx.x, tid = threadIdx.x;
  float c = fmaxf(cnt[gi], 1.0f);
  g[tid] = pool[(size_t)gi * HIDD + tid] / c;
  __syncthreads();
  float a = fc1b[tid];
  for (int k = 0; k < 128; ++k) a += g[k] * fc1w[k * 128 + tid];
  t1[tid] = fmaxf(a, 0.f);
  __syncthreads();
  if (tid < 64) {
    float a2 = fc2b[tid];
    for (int k = 0; k < 128; ++k) a2 += t1[k] * fc2w[k * 64 + tid];
    t2[tid] = fmaxf(a2, 0.f);
  }
  __syncthreads();
  if (tid == 0) {
    float a3 = fc3b[0];
    for (int k = 0; k < 64; ++k) a3 += t2[k] * fc3w[k];
    out[gi] = a3;
  }
}

// ===========================================================================
extern "C" void kernel_launch(void* const* d_in, const int* in_sizes, int n_in,
                              void* d_out, int out_size, void* d_ws,
                              size_t ws_size, hipStream_t stream) {
  const float* x          = (const float*)d_in[0];
  const float* edge_attr  = (const float*)d_in[1];
  const int*   edge_index = (const int*)d_in[2];
  const int*   batch      = (const int*)d_in[3];
  const float* enc_node_w = (const float*)d_in[4];
  const float* enc_node_b = (const float*)d_in[5];
  const float* enc_edge_w = (const float*)d_in[6];
  const float* enc_edge_b = (const float*)d_in[7];
  const float* gat_lin_w  = (const float*)d_in[8];
  const float* gat_att_src  = (const float*)d_in[9];
  const float* gat_att_dst  = (const float*)d_in[10];
  const float* gat_lin_e_w  = (const float*)d_in[11];
  const float* gat_att_edge = (const float*)d_in[12];
  const float* gat_bias   = (const float*)d_in[13];
  const float* bn_gamma   = (const float*)d_in[14];
  const float* bn_beta    = (const float*)d_in[15];
  const float* fc1_w = (const float*)d_in[16];
  const float* fc1_b = (const float*)d_in[17];
  const float* fc2_w = (const float*)d_in[18];
  const float* fc2_b = (const float*)d_in[19];
  const float* fc3_w = (const float*)d_in[20];
  const float* fc3_b = (const float*)d_in[21];
  float* out = (float*)d_out;

  const int* srcI = edge_index;        // row 0
  const int* dstI = edge_index + EE;   // row 1

  // -------- workspace carve-out (floats, 256B-aligned chunks) -------------
  float* base = (float*)d_ws;
  size_t off = 0;
  auto alloc = [&](size_t count) {
    float* r = base + off;
    off += (count + 63) & ~(size_t)63;
    return r;
  };
  float* h       = alloc((size_t)NN * HIDD);
  float* hn      = alloc((size_t)NN * HIDD);
  float* xp      = alloc((size_t)NN * HIDD);
  float* sbuf    = alloc((size_t)NN * 4);
  float* dbuf    = alloc((size_t)NN * 4);
  float* e_real  = alloc((size_t)EE * 16);
  float* e_loop  = alloc((size_t)NN * 16);
  float* attrsum = alloc((size_t)NN * 3);
  float* degb    = alloc((size_t)NN);
  float* mbuf    = alloc((size_t)NN * 4);   // used as unsigned
  float* zbuf    = alloc((size_t)NN * 4);
  float* wbuf    = alloc((size_t)ET * 4);
  float* W2      = alloc(48);
  float* b2      = alloc(16);
  float* bnsums  = alloc(256);
  float* poolb   = alloc((size_t)GG * HIDD);
  float* cntb    = alloc((size_t)GG);
  (void)ws_size; (void)n_in; (void)in_sizes; (void)out_size;

  // -------- one-time (per call) preprocessing -----------------------------
  hipMemsetAsync(attrsum, 0, (size_t)NN * 3 * sizeof(float), stream);
  hipMemsetAsync(degb, 0, (size_t)NN * sizeof(float), stream);

  enc_node_kernel<<<(NN * HIDD + 255) / 256, 256, 0, stream>>>(
      x, enc_node_w, enc_node_b, h);
  edge_prep_kernel<<<(EE + 255) / 256, 256, 0, stream>>>(
      dstI, edge_attr, attrsum, degb);
  weff_kernel<<<1, 256, 0, stream>>>(gat_lin_e_w, gat_att_edge, enc_edge_w,
                                     enc_edge_b, W2, b2);
  e_real_kernel<<<(EE + 255) / 256, 256, 0, stream>>>(edge_attr, W2, b2,
                                                      e_real);
  e_loop_kernel<<<(NN + 255) / 256, 256, 0, stream>>>(attrsum, degb, W2, b2,
                                                      e_loop);

  // -------- 4 GAT layers --------------------------------------------------
  for (int layer = 0; layer < 4; ++layer) {
    const float* linW  = gat_lin_w + (size_t)layer * HIDD * HIDD;
    const float* asrc  = gat_att_src + (size_t)layer * 128;
    const float* adst  = gat_att_dst + (size_t)layer * 128;
    const float* biasL = gat_bias + (size_t)layer * HIDD;
    const float* gamL  = bn_gamma + (size_t)layer * HIDD;
    const float* betL  = bn_beta + (size_t)layer * HIDD;

    xp_gemm_wmma<<<NN / 16, 256, 0, stream>>>(h, linW, xp);
    sd_kernel<<<(NN * 4 + 255) / 256, 256, 0, stream>>>(xp, asrc, adst, sbuf,
                                                        dbuf);

    hipMemsetAsync(mbuf, 0, (size_t)NN * 4 * sizeof(float), stream); // 0 == mapped min
    hipMemsetAsync(zbuf, 0, (size_t)NN * 4 * sizeof(float), stream);
    hipMemsetAsync(hn, 0, (size_t)NN * HIDD * sizeof(float), stream);

    att_pass1<<<(ET + 255) / 256, 256, 0, stream>>>(
        srcI, dstI, sbuf, dbuf, e_real, e_loop, (unsigned*)mbuf, layer * 4);
    att_pass2<<<(ET + 255) / 256, 256, 0, stream>>>(
        srcI, dstI, sbuf, dbuf, e_real, e_loop, (const unsigned*)mbuf, wbuf,
        zbuf, layer * 4);
    att_pass3<<<(ET + 1) / 2, 256, 0, stream>>>(srcI, dstI, xp, wbuf, zbuf,
                                                hn);

    hipMemsetAsync(bnsums, 0, 256 * sizeof(float), stream);
    bn_stats<<<256, 256, 0, stream>>>(hn, biasL, bnsums);
    bn_apply<<<(NN * HIDD + 255) / 256, 256, 0, stream>>>(
        hn, biasL, bnsums, gamL, betL, h, layer);
  }

  // -------- readout -------------------------------------------------------
  hipMemsetAsync(poolb, 0, (size_t)GG * HIDD * sizeof(float), stream);
  hipMemsetAsync(cntb, 0, (size_t)GG * sizeof(float), stream);
  pool_kernel<<<(NN * HIDD + 255) / 256, 256, 0, stream>>>(h, batch, poolb,
                                                           cntb);
  head_kernel<<<GG, 128, 0, stream>>>(poolb, cntb, fc1_w, fc1_b, fc2_w, fc2_b,
                                      fc3_w, fc3_b, out);
}